// Gate_66030827209031
// MI455X (gfx1250) — compile-verified
//
#include <hip/hip_runtime.h>
#include <hip/hip_bf16.h>
#include <stdint.h>

typedef float v2f __attribute__((ext_vector_type(2)));
typedef float v8f __attribute__((ext_vector_type(8)));

#define N_TOKENS 32768
#define DIM      4096
#define N_EXPERTS 64
#define TOPK      2
#define BT        64            // tokens per block (4 token-groups of 16)
#define BK        64            // K tile per stage
#define LDA       68            // padded LDS row stride: banks 4r+k -> conflict-free b64 frag reads
#define LDS_SC    65            // padded score stride: bank (m+e)%64 -> conflict-free epilogue
#define NK       (DIM / BK)     // 64 K-steps
#define NWG      (N_TOKENS / BT)  // 512 workgroups

__device__ __forceinline__ float4 nanfix4(float4 v) {
    v.x = (v.x == v.x) ? v.x : 0.0f;
    v.y = (v.y == v.y) ? v.y : 0.0f;
    v.z = (v.z == v.z) ? v.z : 0.0f;
    v.w = (v.w == v.w) ? v.w : 0.0f;
    return v;
}

// Weight tile stage: per-lane 2x16B async DMA global->LDS, ASYNCcnt-tracked.
// ISA: the instruction offset is added to both the LDS address and the global address,
// so offset:16 advances BOTH sides to the second float4.
__device__ __forceinline__ void async_w_tile(uint32_t lds_addr, uint32_t voff, uint64_t base) {
    asm volatile("global_load_async_to_lds_b128 %0, %1, %2\n\t"
                 "global_load_async_to_lds_b128 %0, %1, %2 offset:16"
                 :: "v"(lds_addr), "v"(voff), "s"(base) : "memory");
}
__device__ __forceinline__ void wait_async0() {
    asm volatile("s_wait_asynccnt 0x0" ::: "memory");
}

__global__ __launch_bounds__(512)
void gate_gemm_top2(const float* __restrict__ x,
                    const float* __restrict__ w,
                    const float* __restrict__ bias,
                    float* __restrict__ out,
                    float* __restrict__ partials)
{
    __shared__ float XS[2][BT * LDA];         // x tiles (double buffered): 64 tokens x 64 K
    __shared__ float WS[2][N_EXPERTS * LDA];  // weight tiles (double buffered): 64 experts x 64 K
    __shared__ float SC[BT * LDS_SC];         // scores: 64 tokens x 64 experts
    __shared__ float TW[BT * TOPK];
    __shared__ int   TI[BT * TOPK];

    const int t    = threadIdx.x;
    const int wave = t >> 5;
    const int lane = t & 31;
    const int ln   = lane & 15;      // row within 16-block
    const int hl   = lane >> 4;      // K-pair selector (0/1)
    const int tg   = wave >> 2;      // token group 0..3
    const int eg   = wave & 3;       // expert group 0..3
    const int m0   = tg * 16;
    const int n0   = eg * 16;
    const int tok0 = blockIdx.x * BT;

    // staging map: 512 threads, thread t owns row t/8 (0..63), 8 floats at col (t%8)*8
    const int sr   = t >> 3;
    const int scol = (t & 7) * 8;

    const float* xg = x + (size_t)(tok0 + sr) * DIM + scol;   // + k per stage
    const uint64_t wbase = (uint64_t)(uintptr_t)w;
    const uint32_t wvoff = (uint32_t)((sr * DIM + scol) * sizeof(float));
    // addrspace(3) offset == low 32 bits of the generic LDS pointer
    uint32_t wlds0 = (uint32_t)(uintptr_t)&WS[0][sr * LDA + scol];
    uint32_t wlds1 = (uint32_t)(uintptr_t)&WS[1][sr * LDA + scol];
    float* xs0 = &XS[0][sr * LDA + scol];
    float* xs1 = &XS[1][sr * LDA + scol];

    v8f c = {};

    // ---- prologue: stage tile 0 into buffer 0 ----
    float4 xa0 = nanfix4(*(const float4*)(xg));
    float4 xa1 = nanfix4(*(const float4*)(xg + 4));
    *(float4*)xs0       = xa0;
    *(float4*)(xs0 + 4) = xa1;
    async_w_tile(wlds0, wvoff, wbase);
    wait_async0();
    __syncthreads();

    for (int kt = 0; kt < NK; ++kt) {
        const int  cur  = kt & 1;
        const bool more = (kt + 1) < NK;
        if (more) {
            const int k1 = (kt + 1) * BK;
            // issue next x tile (through regs for nan_to_num) and next weight tile
            // (async DMA into the ping buffer; its old readers were fenced by the
            // barrier that ended the previous iteration)
            xa0 = nanfix4(*(const float4*)(xg + k1));
            xa1 = nanfix4(*(const float4*)(xg + k1 + 4));
            async_w_tile(cur ? wlds0 : wlds1,
                         wvoff + (uint32_t)(k1 * sizeof(float)), wbase);
        }

        // ---- 16x fp32 WMMA on the current tile ----
        // A frag: lane ln -> x[tok0+m0+ln][k + 2*hl + {0,1}]
        // B frag: lane ln -> W[n0+ln][k + 2*hl + {0,1}]
        const float* ap = &XS[cur][(m0 + ln) * LDA + 2 * hl];
        const float* bp = &WS[cur][(n0 + ln) * LDA + 2 * hl];
#pragma unroll
        for (int kk = 0; kk < BK / 4; ++kk) {
            v2f a = *(const v2f*)(ap + 4 * kk);
            v2f b = *(const v2f*)(bp + 4 * kk);
            c = __builtin_amdgcn_wmma_f32_16x16x4_f32(
                    false, a, false, b, (short)0, c, false, false);
        }

        if (more) {
            float* xsN = cur ? xs0 : xs1;
            *(float4*)xsN       = xa0;
            *(float4*)(xsN + 4) = xa1;
            wait_async0();       // this wave's weight DMA landed; barrier covers the rest
        }
        __syncthreads();
    }

    // ---- scatter C frags: C[r] is (token m0 + r + 8*hl, expert n0 + ln) ----
#pragma unroll
    for (int r = 0; r < 8; ++r)
        SC[(m0 + r + 8 * hl) * LDS_SC + n0 + ln] = c[r];
    __syncthreads();

    // ---- per-token top-2; softmax denominator cancels in topv/sum(topv) ----
    if (t < BT) {
        const int m = t;
        float v1 = -__builtin_inff(), v2 = -__builtin_inff();
        int   i1 = 0, i2 = 0;
        for (int e = 0; e < N_EXPERTS; ++e) {
            float s = SC[m * LDS_SC + e] + bias[e];
            if (s > v1)      { v2 = v1; i2 = i1; v1 = s; i1 = e; }
            else if (s > v2) { v2 = s;  i2 = e; }
        }
        float e2  = expf(v2 - v1);
        float inv = 1.0f / (1.0f + e2);
        float w1  = inv;            // ROUTE_SCALE == 1.0
        float w2  = e2 * inv;
        const int tok = tok0 + m;
        out[tok * 2 + 0] = w1;
        out[tok * 2 + 1] = w2;
        out[(size_t)N_TOKENS * 2 + tok * 2 + 0] = (float)i1;
        out[(size_t)N_TOKENS * 2 + tok * 2 + 1] = (float)i2;
        TW[m * 2] = w1; TW[m * 2 + 1] = w2;
        TI[m * 2] = i1; TI[m * 2 + 1] = i2;
    }
    __syncthreads();

    // ---- deterministic per-block expert sums (fixed-order scan, no float atomics) ----
    if (t < N_EXPERTS) {
        float s = 0.0f;
        for (int j = 0; j < BT * TOPK; ++j)
            s += (TI[j] == t) ? TW[j] : 0.0f;
        partials[(size_t)blockIdx.x * N_EXPERTS + t] = s;
    }
}

__global__ __launch_bounds__(256)
void gate_bias_update(const float* __restrict__ partials,
                      const float* __restrict__ bias,
                      const float* __restrict__ target,
                      float* __restrict__ out)
{
    __shared__ float P[256];
    __shared__ float ES[N_EXPERTS];
    const int t = threadIdx.x;

    // index i = wg*64 + e; i % 64 == t % 64: each thread owns a fixed
    // (expert, chunk) pair and sums in fixed order -> bitwise deterministic.
    float s = 0.0f;
    const int total = NWG * N_EXPERTS;   // 32768, fully coalesced stride-256
    for (int i = t; i < total; i += 256) s += partials[i];
    P[t] = s;
    __syncthreads();

    if (t < N_EXPERTS)
        ES[t] = P[t] + P[t + 64] + P[t + 128] + P[t + 192];
    __syncthreads();

    if (t < N_EXPERTS) {
        float tot = 0.0f;
        for (int e = 0; e < N_EXPERTS; ++e) tot += ES[e];
        float nb = bias[t] + 0.001f * (target[t] * tot - ES[t]) / tot;
        out[(size_t)N_TOKENS * 4 + t] = nb;
    }
}

extern "C" void kernel_launch(void* const* d_in, const int* in_sizes, int n_in,
                              void* d_out, int out_size, void* d_ws, size_t ws_size,
                              hipStream_t stream) {
    const float* x      = (const float*)d_in[0];
    const float* weight = (const float*)d_in[1];
    const float* bias   = (const float*)d_in[2];
    const float* target = (const float*)d_in[3];
    float* out      = (float*)d_out;
    float* partials = (float*)d_ws;   // needs NWG*64*4 = 128 KB scratch

    gate_gemm_top2<<<NWG, 512, 0, stream>>>(x, weight, bias, out, partials);
    gate_bias_update<<<1, 256, 0, stream>>>(partials, bias, target, out);
}